// RetraceQLoss_33036888440948
// MI455X (gfx1250) — compile-verified
//
#include <hip/hip_runtime.h>
#include <stdint.h>

// ---------------------------------------------------------------------------
// Retrace(lambda) Q loss, MI455X / gfx1250, wave32.
// q[t] = a[t] + b[t]*q[t+1]  (backward affine recurrence per row)
//   c  = 0.95*exp(min(lp-olp,0));  m = !timeout
//   b  = pcont*m*c
//   a  = rewards + pcont*((tqv+entropy) - m*c*tq1)
// loss = mean((current_qf - q)^2)
// Bandwidth-bound: ~232MB @ 23.3TB/s ~= 10us floor. Single-pass streaming,
// wave-per-row affine suffix scan, async global->LDS staging of current_qf.
// ---------------------------------------------------------------------------

#define T_LEN    2048
#define SEG      64            // time steps per lane (T_LEN / 32)
#define CHUNKS   16            // SEG / 4 (float4 chunks)
#define WAVES    2             // waves (rows) per block
#define LANES    32
#define PAD_SEG  68            // padded lane stride in LDS (16B aligned, 2-way banks)
#define LAMBDA_F 0.95f

__device__ __forceinline__ void ab_elem(float tq1, float tqv, float rw, float pc,
                                        float lp, float olp, unsigned char tm,
                                        float ent, float& a, float& b)
{
    float c  = LAMBDA_F * __expf(fminf(lp - olp, 0.0f));
    float m  = tm ? 0.0f : 1.0f;
    float mc = m * c;
    b = pc * mc;
    a = rw + pc * ((tqv + ent) - mc * tq1);
}

__global__ __launch_bounds__(WAVES * LANES)
void retrace_row_kernel(const float* __restrict__ current_qf,
                        const float* __restrict__ target_qf_tp1,
                        const float* __restrict__ target_q_values_tp1,
                        const float* __restrict__ rewards,
                        const float* __restrict__ pcont,
                        const float* __restrict__ logpacs,
                        const float* __restrict__ old_logpacs,
                        const float* __restrict__ entropy,
                        const unsigned char* __restrict__ timeout,
                        float* __restrict__ row_partials,
                        int nrows)
{
    __shared__ __align__(16) float a_lds[WAVES][LANES * PAD_SEG];
    __shared__ __align__(16) float b_lds[WAVES][LANES * PAD_SEG];
    __shared__ __align__(16) float c_lds[WAVES][LANES * PAD_SEG];

    const int lane = threadIdx.x & 31;
    const int wv   = threadIdx.x >> 5;
    const int row  = blockIdx.x * WAVES + wv;
    if (row >= nrows) return;

    const long long rb = (long long)row * T_LEN;

    // ---- Stage current_qf row into LDS with async copies (gfx1250 path). ---
    // Global side: fully coalesced 512B per issue. LDS side: padded layout
    // c_lds[wv][owner*PAD_SEG + ii] so the replay pass reads conflict-light.
    {
        const float* gptr = current_qf + rb;
        unsigned cbase = (unsigned)(uintptr_t)(&c_lds[wv][0]);
        #pragma unroll
        for (int j = 0; j < CHUNKS; ++j) {
            int e        = j * 128 + lane * 4;      // element index in row
            int owner    = e >> 6;                  // owning lane's segment
            int ii       = e & 63;
            unsigned dst = cbase + (unsigned)((owner * PAD_SEG + ii) * 4);
            unsigned off = (unsigned)(e * 4);       // byte offset from gptr
            asm volatile("global_load_async_to_lds_b128 %0, %1, %2"
                         :: "v"(dst), "v"(off), "s"(gptr) : "memory");
        }
    }

    const float ent = entropy[0];
    const int   s0  = lane * SEG;   // first t of this lane's segment

    const float4* p_tq1 = (const float4*)(target_qf_tp1       + rb + s0);
    const float4* p_tqv = (const float4*)(target_q_values_tp1 + rb + s0);
    const float4* p_rw  = (const float4*)(rewards             + rb + s0);
    const float4* p_pc  = (const float4*)(pcont               + rb + s0);
    const float4* p_lp  = (const float4*)(logpacs             + rb + s0);
    const float4* p_olp = (const float4*)(old_logpacs         + rb + s0);
    const uchar4* p_to  = (const uchar4*)(timeout             + rb + s0);

    // ---- Pass A: stream inputs, build (a,b), compose lane's affine map. ---
    // Composed map (A,B): q[s0] = A + B * q[s0 + SEG], built in descending t.
    float A = 0.0f, Bc = 1.0f;
    for (int j = CHUNKS - 1; j >= 0; --j) {
        float4 tq1 = p_tq1[j];
        float4 tqv = p_tqv[j];
        float4 rw  = p_rw[j];
        float4 pc  = p_pc[j];
        float4 lp  = p_lp[j];
        float4 olp = p_olp[j];
        uchar4 tm  = p_to[j];

        float av[4], bv[4];
        ab_elem(tq1.x, tqv.x, rw.x, pc.x, lp.x, olp.x, tm.x, ent, av[0], bv[0]);
        ab_elem(tq1.y, tqv.y, rw.y, pc.y, lp.y, olp.y, tm.y, ent, av[1], bv[1]);
        ab_elem(tq1.z, tqv.z, rw.z, pc.z, lp.z, olp.z, tm.z, ent, av[2], bv[2]);
        ab_elem(tq1.w, tqv.w, rw.w, pc.w, lp.w, olp.w, tm.w, ent, av[3], bv[3]);

        // Final time step: q[T-1] = target_qf_tp1[:,T-1], no correction.
        if (lane == 31 && j == CHUNKS - 1) { av[3] = tq1.w; bv[3] = 0.0f; }

        #pragma unroll
        for (int k = 3; k >= 0; --k) {      // descending t within chunk
            A  = fmaf(bv[k], A, av[k]);
            Bc = Bc * bv[k];
        }

        const int lb = lane * PAD_SEG + j * 4;
        a_lds[wv][lb + 0] = av[0]; a_lds[wv][lb + 1] = av[1];
        a_lds[wv][lb + 2] = av[2]; a_lds[wv][lb + 3] = av[3];
        b_lds[wv][lb + 0] = bv[0]; b_lds[wv][lb + 1] = bv[1];
        b_lds[wv][lb + 2] = bv[2]; b_lds[wv][lb + 3] = bv[3];
    }

    // ---- Wave-level suffix scan of affine maps (Hillis-Steele, 5 steps). ---
    // P_L = g_L o g_{L+1} o ... o g_31 ; combine: (g o h): A+=B*Ah, B*=Bh.
    #pragma unroll
    for (int d = 1; d < 32; d <<= 1) {
        float An = __shfl_down(A,  d, 32);
        float Bn = __shfl_down(Bc, d, 32);
        if (lane + d < 32) { A = fmaf(Bc, An, A); Bc = Bc * Bn; }
    }
    // Exclusive carry for lane L = P_{L+1}(0)  (B of suffix incl. lane31 is 0).
    float Ae    = __shfl_down(A, 1, 32);
    float carry = (lane == 31) ? 0.0f : Ae;

    // ---- Pass B: replay recurrence from LDS, accumulate squared error. ----
    asm volatile("s_wait_asynccnt 0x0" ::: "memory");  // current_qf staged

    float q   = carry;
    float acc = 0.0f;
    for (int j = CHUNKS - 1; j >= 0; --j) {
        #pragma unroll
        for (int k = 3; k >= 0; --k) {
            const int idx = lane * PAD_SEG + j * 4 + k;
            q = fmaf(b_lds[wv][idx], q, a_lds[wv][idx]);
            float d = c_lds[wv][idx] - q;
            acc = fmaf(d, d, acc);
        }
    }

    // ---- Wave reduction (fixed xor tree -> deterministic). -----------------
    #pragma unroll
    for (int d = 16; d >= 1; d >>= 1) acc += __shfl_xor(acc, d, 32);
    if (lane == 0) row_partials[row] = acc;
}

// Deterministic final reduction: fixed-order strided sums + LDS tree.
__global__ __launch_bounds__(256)
void retrace_reduce_kernel(const float* __restrict__ partials, int n,
                           float inv_count, float* __restrict__ out)
{
    __shared__ float s[256];
    float acc = 0.0f;
    for (int i = threadIdx.x; i < n; i += 256) acc += partials[i];
    s[threadIdx.x] = acc;
    __syncthreads();
    for (int step = 128; step >= 1; step >>= 1) {
        if ((int)threadIdx.x < step) s[threadIdx.x] += s[threadIdx.x + step];
        __syncthreads();
    }
    if (threadIdx.x == 0) out[0] = s[0] * inv_count;
}

extern "C" void kernel_launch(void* const* d_in, const int* in_sizes, int n_in,
                              void* d_out, int out_size, void* d_ws, size_t ws_size,
                              hipStream_t stream)
{
    const float*         current_qf  = (const float*)d_in[0];
    const float*         tq1         = (const float*)d_in[1];
    const float*         tqv         = (const float*)d_in[2];
    const float*         rewards     = (const float*)d_in[3];
    const float*         pcont       = (const float*)d_in[4];
    const float*         logpacs     = (const float*)d_in[5];
    const float*         old_logpacs = (const float*)d_in[6];
    const float*         entropy     = (const float*)d_in[7];
    const unsigned char* timeout     = (const unsigned char*)d_in[8];

    const int BT    = in_sizes[0];
    const int nrows = BT / T_LEN;

    float* partials = (float*)d_ws;

    dim3 grid((nrows + WAVES - 1) / WAVES);
    retrace_row_kernel<<<grid, WAVES * LANES, 0, stream>>>(
        current_qf, tq1, tqv, rewards, pcont, logpacs, old_logpacs,
        entropy, timeout, partials, nrows);

    retrace_reduce_kernel<<<1, 256, 0, stream>>>(
        partials, nrows, 1.0f / (float)BT, (float*)d_out);
}